// TokenEmbedding_74646531604771
// MI455X (gfx1250) — compile-verified
//
#include <hip/hip_runtime.h>

typedef __attribute__((ext_vector_type(2))) float v2f;
typedef __attribute__((ext_vector_type(8))) float v8f;

#define DIM 64
#define MAT (DIM * DIM)   // 4096 floats = 16 KB

// ---------------------------------------------------------------------------
// One 16-row band (rows [mt*16, mt*16+16)) of a 64x64x64 f32 GEMM:
//   lC[band] = lA[band] @ lB      (row-major, stride DIM, LDS-resident)
// 4 C tiles x 16 K-steps = 64 x V_WMMA_F32_16X16X4_F32 per call.
// VGPR layouts per CDNA5 ISA 7.12.2:
//   A 16x4 f32 : lanes 0-15 -> {v0=K0, v1=K1}, lanes 16-31 -> {v0=K2, v1=K3}
//   B 4x16 f32 : v0 = rows {K0 | K2}, v1 = rows {K1 | K3}, N = lane&15
//   C 16x16 f32: vr = row {r | r+8}, N = lane&15
// Requires EXEC all-1s: callers keep control flow uniform.
// ---------------------------------------------------------------------------
__device__ __forceinline__ void wave_gemm64_band(const float* lA, const float* lB,
                                                 float* lC, int lane, int mt) {
    const int ln    = lane & 15;
    const int khalf = (lane >> 4) * 2;   // 0 or 2
    const int chalf = (lane >> 4) * 8;   // 0 or 8
    const int m     = mt * 16 + ln;
    #pragma unroll
    for (int nt = 0; nt < 4; ++nt) {
        const int n = nt * 16 + ln;
        v8f c = {0.f, 0.f, 0.f, 0.f, 0.f, 0.f, 0.f, 0.f};
        #pragma unroll
        for (int kt = 0; kt < 16; ++kt) {
            const int k0 = kt * 4 + khalf;
            v2f a, b;
            a[0] = lA[m * DIM + k0];
            a[1] = lA[m * DIM + k0 + 1];
            b[0] = lB[k0 * DIM + n];
            b[1] = lB[(k0 + 1) * DIM + n];
            c = __builtin_amdgcn_wmma_f32_16x16x4_f32(
                    false, a, false, b, (short)0, c, false, false);
        }
        #pragma unroll
        for (int r = 0; r < 8; ++r)
            lC[(mt * 16 + chalf + r) * DIM + n] = c[r];
    }
}

// ---------------------------------------------------------------------------
// Kernel 1: Q_g = expm(P_g^T - P_g) = prims[g]^T   (g = 0,1)
// Scaling-and-squaring: X = (P^T - P)/2^8, 10-term Taylor via Horner,
// then 8 squarings.  One wave per generator, everything in LDS.
// ---------------------------------------------------------------------------
__global__ __launch_bounds__(32)
void expm_qt_kernel(const float* __restrict__ prims_raw, float* __restrict__ qout) {
    __shared__ float X[MAT];
    __shared__ float Buf[2][MAT];

    const int g    = blockIdx.x;
    const int lane = threadIdx.x;
    const float* P = prims_raw + g * MAT;

    // X = (P^T - P) * 2^-8 ;  Buf[0] = I
    for (int i = lane; i < MAT; i += 32) {
        const int r = i >> 6, c = i & 63;
        X[i]      = (P[c * DIM + r] - P[r * DIM + c]) * (1.0f / 256.0f);
        Buf[0][i] = (r == c) ? 1.0f : 0.0f;
    }
    __syncthreads();

    // Horner: R = I + X/k * R, k = 10..1
    int cur = 0;
    for (int k = 10; k >= 1; --k) {
        #pragma unroll
        for (int mt = 0; mt < 4; ++mt)
            wave_gemm64_band(X, Buf[cur], Buf[cur ^ 1], lane, mt);  // tmp = X @ R
        cur ^= 1;
        __syncthreads();
        const float inv = 1.0f / (float)k;
        for (int i = lane; i < MAT; i += 32) {
            const int r = i >> 6, c = i & 63;
            Buf[cur][i] = Buf[cur][i] * inv + ((r == c) ? 1.0f : 0.0f);
        }
        __syncthreads();
    }

    // undo scaling: 8 squarings
    for (int s = 0; s < 8; ++s) {
        #pragma unroll
        for (int mt = 0; mt < 4; ++mt)
            wave_gemm64_band(Buf[cur], Buf[cur], Buf[cur ^ 1], lane, mt);
        cur ^= 1;
        __syncthreads();
    }

    for (int i = lane; i < MAT; i += 32)
        qout[g * MAT + i] = Buf[cur][i];
}

// ---------------------------------------------------------------------------
// Kernel 2: per-token positional map  M = prod over bits of Q_{bit}
//   (mirrors reference: M <- M @ prims[bit]^T == M @ Q_bit, LSB first,
//    while pos > 1), plus the content-embedding gather.
// 128 threads = 4 waves per token: each wave owns one 16-row C band, so the
// dependent GEMM chain's critical path is 4x shorter and the WGP runs 20
// waves (5 blocks x 4) instead of 5.  Q is staged global->LDS with
// GLOBAL_LOAD_ASYNC_TO_LDS_B128 (ASYNCcnt), overlapped with the M=I init.
// ---------------------------------------------------------------------------
__global__ __launch_bounds__(128)
void maps_content_kernel(const int* __restrict__ tt, const int* __restrict__ tv,
                         const int* __restrict__ np,
                         const float* __restrict__ embed,
                         const float* __restrict__ q,
                         float* __restrict__ content, float* __restrict__ maps,
                         int ntok) {
    __shared__ float Qs[2 * MAT];
    __shared__ float Buf[2][MAT];

    const int tid  = threadIdx.x;
    const int lane = tid & 31;
    const int wid  = tid >> 5;
    const int blk  = blockIdx.x;
    if (blk >= ntok) return;   // uniform per block

    // --- async-stage Q0,Q1 (32 KB) into LDS: 16B per lane per op, GV mode ---
    {
        const uint32_t lds_base = (uint32_t)(uintptr_t)(&Qs[0]); // addr[31:0] = LDS offset
        #pragma unroll
        for (int i = tid; i < (2 * MAT) / 4; i += 128) {
            const uint64_t ga = (uint64_t)(uintptr_t)(q + 4 * i);
            const uint32_t la = lds_base + 16u * (uint32_t)i;
            asm volatile("global_load_async_to_lds_b128 %0, %1, off"
                         :: "v"(la), "v"(ga) : "memory");
        }
    }

    // overlap: init M = I while the async DMA runs
    for (int i = tid; i < MAT; i += 128)
        Buf[0][i] = ((i >> 6) == (i & 63)) ? 1.0f : 0.0f;

    asm volatile("s_wait_asynccnt 0" ::: "memory");
    __syncthreads();

    int pos = __builtin_amdgcn_readfirstlane(np[blk]);  // scalar -> uniform branches
    int cur = 0;
    while (pos > 1) {
        const int b = pos & 1;
        wave_gemm64_band(Buf[cur], Qs + b * MAT, Buf[cur ^ 1], lane, wid);
        cur ^= 1;
        pos >>= 1;
        __syncthreads();
    }

    // stream the 16 KB map out with 128-bit stores
    float4*       dst = (float4*)(maps + (size_t)blk * MAT);
    const float4* src = (const float4*)Buf[cur];
    for (int i = tid; i < MAT / 4; i += 128) dst[i] = src[i];

    // content embedding gather (reference's where-chain precedence)
    const int ty = tt[blk];
    const int va = tv[blk];
    int  idx   = 0;
    bool valid = false;
    if (ty == 0)                    { idx = 0;       valid = true; }
    else if (ty == 1)               { idx = va + 1;  valid = true; }
    else if (ty == 2)               { idx = va + 5;  valid = true; }
    else if (ty == 4)               { idx = 8;       valid = true; }
    else if (ty == 3 && va == -1)   { idx = 10;      valid = true; }

    float* cdst = content + (size_t)blk * DIM;
    for (int i = tid; i < DIM; i += 128)
        cdst[i] = valid ? embed[idx * DIM + i] : 0.0f;
}

// ---------------------------------------------------------------------------
extern "C" void kernel_launch(void* const* d_in, const int* in_sizes, int n_in,
                              void* d_out, int out_size, void* d_ws, size_t ws_size,
                              hipStream_t stream) {
    const int*   token_types  = (const int*)d_in[0];
    const int*   token_values = (const int*)d_in[1];
    const int*   node_pos     = (const int*)d_in[2];
    const float* embed_table  = (const float*)d_in[3];
    const float* prims_raw    = (const float*)d_in[4];

    const int ntok = in_sizes[0];              // B*S = 4096

    float* q       = (float*)d_ws;             // 2 * 64*64 f32 = 32 KB scratch
    float* content = (float*)d_out;            // [ntok, 64]
    float* maps    = content + (size_t)ntok * DIM;  // [ntok, 64, 64]

    expm_qt_kernel<<<2, 32, 0, stream>>>(prims_raw, q);
    maps_content_kernel<<<ntok, 128, 0, stream>>>(token_types, token_values,
                                                  node_pos, embed_table, q,
                                                  content, maps, ntok);
}